// BioEncoder_85667417686130
// MI455X (gfx1250) — compile-verified
//
#include <hip/hip_runtime.h>

// ---------------------------------------------------------------- constants
#define N_NODES   50000
#define N_EDGES   800000
#define F_IN      78
#define OUTF      256
#define N_GRAPHS  1024
#define B_MIC     4096
#define D_MIC     1000
#define B_DIS     4096
#define D_DIS     500
#define BN_EPS    1e-5f

typedef float v2f __attribute__((ext_vector_type(2)));
typedef float v8f __attribute__((ext_vector_type(8)));

// ---------------------------------------------------------------- GEMM (WMMA f32 16x16x4)
// C[M,N] = optional_relu(A[M,K] @ W[K,N] + bias[N])
// One wave computes a 16x64 tile of C (4 accumulators) so each A fragment
// feeds 4 WMMAs per K-step. M % 16 == 0 and N % 64 == 0 at all call sites;
// K may be ragged (K=78 handled by a guarded tail step).
__global__ __launch_bounds__(256)
void gemm_bias_wmma(const float* __restrict__ A, const float* __restrict__ W,
                    const float* __restrict__ bias, float* __restrict__ C,
                    int M, int K, int N, int do_relu)
{
    const int wave = threadIdx.x >> 5;
    const int lane = threadIdx.x & 31;
    const int half = lane >> 4;     // selects K upper pair (A/B) / M upper half (C)
    const int lid  = lane & 15;

    const int tiles_n = N >> 6;     // 64-wide N tiles
    const int tiles_m = M >> 4;
    const int wid = blockIdx.x * (blockDim.x >> 5) + wave;
    const int tm  = wid / tiles_n;
    const int tn  = wid - tm * tiles_n;
    if (tm >= tiles_m) return;      // wave-uniform: EXEC stays all-ones for WMMA

    const int arow = tm * 16 + lid; // A row owned by this lane
    const int col0 = tn * 64 + lid; // first of 4 B/C columns owned by this lane

    const float* __restrict__ Arow = A + (size_t)arow * K;

    v8f acc0 = {}, acc1 = {}, acc2 = {}, acc3 = {};

    const int K4 = K & ~3;
    int k0 = 0;
    for (; k0 < K4; k0 += 4) {
        const int ka = k0 + 2 * half;
        v2f a;
        a.x = Arow[ka];
        a.y = Arow[ka + 1];
        const float* __restrict__ W0 = W + (size_t)ka * N + col0;  // row ka
        const float* __restrict__ W1 = W0 + N;                     // row ka+1
        v2f b0, b1, b2, b3;
        b0.x = W0[0];   b0.y = W1[0];
        b1.x = W0[16];  b1.y = W1[16];
        b2.x = W0[32];  b2.y = W1[32];
        b3.x = W0[48];  b3.y = W1[48];
        acc0 = __builtin_amdgcn_wmma_f32_16x16x4_f32(false, a, false, b0, (short)0, acc0, false, false);
        acc1 = __builtin_amdgcn_wmma_f32_16x16x4_f32(false, a, false, b1, (short)0, acc1, false, false);
        acc2 = __builtin_amdgcn_wmma_f32_16x16x4_f32(false, a, false, b2, (short)0, acc2, false, false);
        acc3 = __builtin_amdgcn_wmma_f32_16x16x4_f32(false, a, false, b3, (short)0, acc3, false, false);
    }
    if (k0 < K) {                   // ragged K tail (e.g. K = 78)
        const int ka = k0 + 2 * half;
        const bool g0 = (ka     < K);
        const bool g1 = (ka + 1 < K);
        v2f a;
        a.x = g0 ? Arow[ka]     : 0.f;
        a.y = g1 ? Arow[ka + 1] : 0.f;
        const float* __restrict__ W0 = W + (size_t)ka * N + col0;
        const float* __restrict__ W1 = W0 + N;
        v2f b0, b1, b2, b3;
        b0.x = g0 ? W0[0]  : 0.f;  b0.y = g1 ? W1[0]  : 0.f;
        b1.x = g0 ? W0[16] : 0.f;  b1.y = g1 ? W1[16] : 0.f;
        b2.x = g0 ? W0[32] : 0.f;  b2.y = g1 ? W1[32] : 0.f;
        b3.x = g0 ? W0[48] : 0.f;  b3.y = g1 ? W1[48] : 0.f;
        acc0 = __builtin_amdgcn_wmma_f32_16x16x4_f32(false, a, false, b0, (short)0, acc0, false, false);
        acc1 = __builtin_amdgcn_wmma_f32_16x16x4_f32(false, a, false, b1, (short)0, acc1, false, false);
        acc2 = __builtin_amdgcn_wmma_f32_16x16x4_f32(false, a, false, b2, (short)0, acc2, false, false);
        acc3 = __builtin_amdgcn_wmma_f32_16x16x4_f32(false, a, false, b3, (short)0, acc3, false, false);
    }

    // Epilogue: C layout per 16x16 tile: VGPR r -> M = r (+8 for hi lanes), N = lane col.
    const float bc0 = bias[col0];
    const float bc1 = bias[col0 + 16];
    const float bc2 = bias[col0 + 32];
    const float bc3 = bias[col0 + 48];
#pragma unroll
    for (int r = 0; r < 8; ++r) {
        const int m = tm * 16 + r + 8 * half;
        float* __restrict__ Crow = C + (size_t)m * N;
        float v0 = acc0[r] + bc0;
        float v1 = acc1[r] + bc1;
        float v2 = acc2[r] + bc2;
        float v3 = acc3[r] + bc3;
        if (do_relu) {
            v0 = fmaxf(v0, 0.f); v1 = fmaxf(v1, 0.f);
            v2 = fmaxf(v2, 0.f); v3 = fmaxf(v3, 0.f);
        }
        Crow[col0]      = v0;
        Crow[col0 + 16] = v1;
        Crow[col0 + 32] = v2;
        Crow[col0 + 48] = v3;
    }
}

// ---------------------------------------------------------------- edge scatter-add
// AGG[dst[e]] += H[src[e]]  (AGG pre-seeded with H via memcpy => h + sum_j h_j)
// One wave per edge, lanes stride the feature row (coalesced atomics).
__global__ __launch_bounds__(256)
void scatter_add_edges(const float* __restrict__ H, const int* __restrict__ src,
                       const int* __restrict__ dst, float* __restrict__ AGG,
                       int n_edges, int F)
{
    const int wid  = (blockIdx.x * blockDim.x + threadIdx.x) >> 5;
    const int lane = threadIdx.x & 31;
    if (wid >= n_edges) return;
    const int s = src[wid];
    const int d = dst[wid];
    const float* hs = H   + (size_t)s * F;
    float*       ad = AGG + (size_t)d * F;
    for (int f = lane; f < F; f += 32)
        atomicAdd(&ad[f], hs[f]);
}

// ---------------------------------------------------------------- segment-sum pool
__global__ __launch_bounds__(256)
void pool_add(const float* __restrict__ H, const int* __restrict__ batch,
              float* __restrict__ P, int n_nodes, int F)
{
    const int wid  = (blockIdx.x * blockDim.x + threadIdx.x) >> 5;
    const int lane = threadIdx.x & 31;
    if (wid >= n_nodes) return;
    const int g = batch[wid];
    const float* hn = H + (size_t)wid * F;
    float*       pg = P + (size_t)g   * F;
    for (int f = lane; f < F; f += 32)
        atomicAdd(&pg[f], hn[f]);
}

// ---------------------------------------------------------------- zero fill
__global__ __launch_bounds__(256)
void zero_fill(float* __restrict__ p, int n)
{
    const int i = blockIdx.x * blockDim.x + threadIdx.x;
    if (i < n) p[i] = 0.f;
}

// ---------------------------------------------------------------- BatchNorm(+ReLU)
// Training-mode BN math (biased variance), one block per column.
__global__ __launch_bounds__(256)
void bn_relu_col(const float* __restrict__ X, const float* __restrict__ gamma,
                 const float* __restrict__ beta, float* __restrict__ Y,
                 int M, int N)
{
    __shared__ float s_sum[256];
    __shared__ float s_sq[256];
    const int j = blockIdx.x;
    float s = 0.f, sq = 0.f;
    for (int i = threadIdx.x; i < M; i += blockDim.x) {
        const float v = X[(size_t)i * N + j];
        s  += v;
        sq += v * v;
    }
    s_sum[threadIdx.x] = s;
    s_sq[threadIdx.x]  = sq;
    __syncthreads();
    for (int st = 128; st > 0; st >>= 1) {
        if (threadIdx.x < st) {
            s_sum[threadIdx.x] += s_sum[threadIdx.x + st];
            s_sq[threadIdx.x]  += s_sq[threadIdx.x + st];
        }
        __syncthreads();
    }
    const float mean = s_sum[0] / (float)M;
    const float var  = s_sq[0] / (float)M - mean * mean;
    const float scl  = rsqrtf(var + BN_EPS) * gamma[j];
    const float sh   = beta[j];
    for (int i = threadIdx.x; i < M; i += blockDim.x) {
        const float v = (X[(size_t)i * N + j] - mean) * scl + sh;
        Y[(size_t)i * N + j] = fmaxf(v, 0.f);
    }
}

// ---------------------------------------------------------------- host driver
static inline int gemm_blocks(int M, int N) { return ((M >> 4) * (N >> 6) + 7) / 8; }

extern "C" void kernel_launch(void* const* d_in, const int* in_sizes, int n_in,
                              void* d_out, int out_size, void* d_ws, size_t ws_size,
                              hipStream_t stream)
{
    // -------- inputs (setup_inputs dict order, params in insertion order)
    const float* x        = (const float*)d_in[0];
    const int*   edge     = (const int*)  d_in[1];  // [2, N_EDGES]
    const int*   batch    = (const int*)  d_in[2];
    const float* mic_feat = (const float*)d_in[3];
    const float* dis_feat = (const float*)d_in[4];

    const float* gin1_W1 = (const float*)d_in[5];
    const float* gin1_b1 = (const float*)d_in[6];
    const float* gin1_W2 = (const float*)d_in[7];
    const float* gin1_b2 = (const float*)d_in[8];
    const float* gin2_W1 = (const float*)d_in[9];
    const float* gin2_b1 = (const float*)d_in[10];
    const float* gin2_W2 = (const float*)d_in[11];
    const float* gin2_b2 = (const float*)d_in[12];
    const float* gin3_W1 = (const float*)d_in[13];
    const float* gin3_b1 = (const float*)d_in[14];
    const float* gin3_W2 = (const float*)d_in[15];
    const float* gin3_b2 = (const float*)d_in[16];
    const float* bn1_g   = (const float*)d_in[17];
    const float* bn1_b   = (const float*)d_in[18];
    const float* bn2_g   = (const float*)d_in[19];
    const float* bn2_b   = (const float*)d_in[20];
    const float* bn3_g   = (const float*)d_in[21];
    const float* bn3_b   = (const float*)d_in[22];
    const float* fc_W    = (const float*)d_in[23];
    const float* fc_b    = (const float*)d_in[24];
    const float* mic_W   = (const float*)d_in[25];
    const float* mic_b   = (const float*)d_in[26];
    const float* mic_g   = (const float*)d_in[27];
    const float* mic_beta= (const float*)d_in[28];
    const float* dis_W   = (const float*)d_in[29];
    const float* dis_b   = (const float*)d_in[30];
    const float* dis_g   = (const float*)d_in[31];
    const float* dis_beta= (const float*)d_in[32];

    const int* e_src = edge;
    const int* e_dst = edge + N_EDGES;

    float* out = (float*)d_out;

    // -------- workspace layout: three 50000x256 fp32 buffers, heavily reused
    const size_t BIG = (size_t)N_NODES * OUTF;          // 12.8M floats
    float* H   = (float*)d_ws;                          // current node features
    float* AGG = H + BIG;                               // agg / 2nd-gemm out / head scratch
    float* T1  = AGG + BIG;                             // mlp hidden / pooled

    const dim3 blk(256);
    const int eb = (N_EDGES + 7) / 8;                   // 8 waves per block
    const int nb = (N_NODES + 7) / 8;

    // ================= GIN layer 1 (K = 78) =================
    hipMemcpyAsync(AGG, x, (size_t)N_NODES * F_IN * sizeof(float),
                   hipMemcpyDeviceToDevice, stream);
    scatter_add_edges<<<eb, blk, 0, stream>>>(x, e_src, e_dst, AGG, N_EDGES, F_IN);
    gemm_bias_wmma<<<gemm_blocks(N_NODES, OUTF), blk, 0, stream>>>(
        AGG, gin1_W1, gin1_b1, T1, N_NODES, F_IN, OUTF, 1);
    gemm_bias_wmma<<<gemm_blocks(N_NODES, OUTF), blk, 0, stream>>>(
        T1, gin1_W2, gin1_b2, AGG, N_NODES, OUTF, OUTF, 0);
    bn_relu_col<<<OUTF, blk, 0, stream>>>(AGG, bn1_g, bn1_b, H, N_NODES, OUTF);

    // ================= GIN layer 2 =================
    hipMemcpyAsync(AGG, H, BIG * sizeof(float), hipMemcpyDeviceToDevice, stream);
    scatter_add_edges<<<eb, blk, 0, stream>>>(H, e_src, e_dst, AGG, N_EDGES, OUTF);
    gemm_bias_wmma<<<gemm_blocks(N_NODES, OUTF), blk, 0, stream>>>(
        AGG, gin2_W1, gin2_b1, T1, N_NODES, OUTF, OUTF, 1);
    gemm_bias_wmma<<<gemm_blocks(N_NODES, OUTF), blk, 0, stream>>>(
        T1, gin2_W2, gin2_b2, AGG, N_NODES, OUTF, OUTF, 0);
    bn_relu_col<<<OUTF, blk, 0, stream>>>(AGG, bn2_g, bn2_b, H, N_NODES, OUTF);

    // ================= GIN layer 3 =================
    hipMemcpyAsync(AGG, H, BIG * sizeof(float), hipMemcpyDeviceToDevice, stream);
    scatter_add_edges<<<eb, blk, 0, stream>>>(H, e_src, e_dst, AGG, N_EDGES, OUTF);
    gemm_bias_wmma<<<gemm_blocks(N_NODES, OUTF), blk, 0, stream>>>(
        AGG, gin3_W1, gin3_b1, T1, N_NODES, OUTF, OUTF, 1);
    gemm_bias_wmma<<<gemm_blocks(N_NODES, OUTF), blk, 0, stream>>>(
        T1, gin3_W2, gin3_b2, AGG, N_NODES, OUTF, OUTF, 0);
    bn_relu_col<<<OUTF, blk, 0, stream>>>(AGG, bn3_g, bn3_b, H, N_NODES, OUTF);

    // ================= global_add_pool + fc head (rows 0..1023) =================
    const int pooledN = N_GRAPHS * OUTF;
    zero_fill<<<(pooledN + 255) / 256, blk, 0, stream>>>(T1, pooledN);
    pool_add<<<nb, blk, 0, stream>>>(H, batch, T1, N_NODES, OUTF);
    gemm_bias_wmma<<<gemm_blocks(N_GRAPHS, OUTF), blk, 0, stream>>>(
        T1, fc_W, fc_b, out, N_GRAPHS, OUTF, OUTF, 1);

    // ================= mic head (rows 1024..5119) =================
    gemm_bias_wmma<<<gemm_blocks(B_MIC, OUTF), blk, 0, stream>>>(
        mic_feat, mic_W, mic_b, AGG, B_MIC, D_MIC, OUTF, 0);
    bn_relu_col<<<OUTF, blk, 0, stream>>>(
        AGG, mic_g, mic_beta, out + (size_t)N_GRAPHS * OUTF, B_MIC, OUTF);

    // ================= dis head (rows 5120..9215) =================
    gemm_bias_wmma<<<gemm_blocks(B_DIS, OUTF), blk, 0, stream>>>(
        dis_feat, dis_W, dis_b, AGG, B_DIS, D_DIS, OUTF, 0);
    bn_relu_col<<<OUTF, blk, 0, stream>>>(
        AGG, dis_g, dis_beta, out + (size_t)(N_GRAPHS + B_MIC) * OUTF, B_DIS, OUTF);

    (void)in_sizes; (void)n_in; (void)out_size; (void)ws_size;
}